// RocketNet_21199958573665
// MI455X (gfx1250) — compile-verified
//
#include <hip/hip_runtime.h>

// ---------------- constants matching the reference ----------------
#define SL      32768
#define NK      2000
#define KMAX    11
#define NS      2
#define TCHUNKS 32
#define MAX_TILES 256

typedef __attribute__((ext_vector_type(2))) float v2f;
typedef __attribute__((ext_vector_type(8))) float v8f;

struct TileDesc {
  int c0_rank;    // global sorted rank of first channel in this tile
  int nch;        // valid channels in tile (1..16)
  int k, d, p;    // conv params of the group
  int Lg;         // output length of the group
  int c_local0;   // local channel offset within group
  int Cg;         // group channel count
  long long base; // group output base offset (elements)
};

// ws layout (ints): [0..15] header (ws[0]=ntiles, ws[1]=total_work)
//                   [16..16+NK) perm (rank -> kernel index)
//                   then TileDesc[MAX_TILES] (8-byte aligned)

// ---------------- numpy legacy MT19937 replication ----------------
__device__ inline unsigned mt_next(unsigned* key, int& pos) {
  if (pos == 624) {
    for (int i = 0; i < 624 - 397; i++) {
      unsigned y = (key[i] & 0x80000000u) | (key[i + 1] & 0x7fffffffu);
      key[i] = key[i + 397] ^ (y >> 1) ^ ((y & 1u) ? 0x9908b0dfu : 0u);
    }
    for (int i = 624 - 397; i < 623; i++) {
      unsigned y = (key[i] & 0x80000000u) | (key[i + 1] & 0x7fffffffu);
      key[i] = key[i + 397 - 624] ^ (y >> 1) ^ ((y & 1u) ? 0x9908b0dfu : 0u);
    }
    unsigned y = (key[623] & 0x80000000u) | (key[0] & 0x7fffffffu);
    key[623] = key[396] ^ (y >> 1) ^ ((y & 1u) ? 0x9908b0dfu : 0u);
    pos = 0;
  }
  unsigned y = key[pos++];
  y ^= (y >> 11);
  y ^= (y << 7)  & 0x9d2c5680u;
  y ^= (y << 15) & 0xefc60000u;
  y ^= (y >> 18);
  return y;
}

__device__ inline double mt_double(unsigned* key, int& pos) {
  unsigned a = mt_next(key, pos) >> 5;
  unsigned b = mt_next(key, pos) >> 6;
  return ((double)a * 67108864.0 + (double)b) / 9007199254740992.0;
}

// randint with inclusive max `rng` (legacy masked rejection, 32-bit fast path)
__device__ inline unsigned mt_randint_incl(unsigned* key, int& pos, unsigned rng) {
  unsigned mask = rng;
  mask |= mask >> 1; mask |= mask >> 2; mask |= mask >> 4;
  mask |= mask >> 8; mask |= mask >> 16;
  unsigned v;
  do { v = mt_next(key, pos) & mask; } while (v > rng);
  return v;
}

// advance stream for `ng` gaussians (legacy polar method; values discarded)
__device__ inline void mt_gauss_skip(unsigned* key, int& pos, bool& has_gauss, int ng) {
  while (ng > 0) {
    if (has_gauss) { has_gauss = false; ng--; continue; }
    double x1, x2, r2;
    do {
      x1 = __dadd_rn(__dmul_rn(2.0, mt_double(key, pos)), -1.0);
      x2 = __dadd_rn(__dmul_rn(2.0, mt_double(key, pos)), -1.0);
      r2 = __dadd_rn(__dmul_rn(x1, x1), __dmul_rn(x2, x2));
    } while (r2 >= 1.0 || r2 == 0.0);
    has_gauss = true;  // spare cached
    ng--;
  }
}

// ---------------- setup: regenerate meta, sort groups, emit tiles ----------------
__global__ __launch_bounds__(256) void rocket_setup(int* __restrict__ ws) {
  __shared__ unsigned s_mt[624];
  __shared__ int s_key[NK];
  const int tid = threadIdx.x;

  if (tid == 0) {
    unsigned seed = 0u;
    for (int i = 0; i < 624; i++) {
      s_mt[i] = seed;
      seed = 1812433253u * (seed ^ (seed >> 30)) + (unsigned)(i + 1);
    }
    int pos = 624;
    bool has_gauss = false;
    for (int i = 0; i < NK; i++) {
      unsigned ks = mt_randint_incl(s_mt, pos, 2u);     // choice of {7,9,11}
      int k = (ks == 0u) ? 7 : ((ks == 1u) ? 9 : 11);
      unsigned e = mt_randint_incl(s_mt, pos, (k == 7) ? 11u : 10u);
      int d = 1 << e;
      double u = mt_double(s_mt, pos);                  // rng.rand()
      int p = (u <= 0.5) ? 0 : ((k - 1) * d) / 2;
      (void)mt_double(s_mt, pos);                       // rng.uniform (bias)
      mt_gauss_skip(s_mt, pos, has_gauss, k);           // rng.normal(size=k)
      s_key[i] = (k << 25) | (d << 13) | p;             // lexicographic (k,d,p)
    }
  }
  __syncthreads();

  for (int i = tid; i < NK; i += 256) {                 // stable rank -> perm
    int ki = s_key[i];
    int r = 0;
    for (int j = 0; j < NK; j++) {
      int kj = s_key[j];
      r += (kj < ki) || (kj == ki && j < i);
    }
    ws[16 + r] = i;
  }
  __syncthreads();

  if (tid == 0) {
    const int* perm = ws + 16;
    TileDesc* tiles = (TileDesc*)(ws + 16 + NK);
    long long base = 0;
    int nt = 0, r = 0;
    while (r < NK) {
      int key = s_key[perm[r]];
      int gs = r;
      while (r < NK && s_key[perm[r]] == key) r++;
      int Cg = r - gs;
      int k = key >> 25;
      int d = (key >> 13) & 0xFFF;
      int p = key & 0x1FFF;
      int Lg = SL + 2 * p - d * (k - 1);
      for (int c0 = 0; c0 < Cg && nt < MAX_TILES; c0 += 16) {
        TileDesc t;
        t.c0_rank = gs + c0;
        t.nch = (Cg - c0 < 16) ? (Cg - c0) : 16;
        t.k = k; t.d = d; t.p = p; t.Lg = Lg;
        t.c_local0 = c0; t.Cg = Cg; t.base = base;
        tiles[nt++] = t;
      }
      base += (long long)NS * Cg * Lg;
    }
    ws[0] = nt;
    ws[1] = nt * NS * TCHUNKS;
  }
}

// ---------------- compute: WMMA f32 16x16x4 dilated conv ----------------
template <int NBLK>
__device__ __attribute__((always_inline)) void conv_run(
    const TileDesc td, const int* __restrict__ perm,
    const float* __restrict__ xs, const float* __restrict__ Wt,
    const float* __restrict__ Bv, float* __restrict__ out,
    long long outbase, int t_begin, int t_end, int lane) {
  const int tcol  = lane & 15;
  const int kband = (lane >> 4) << 1;   // 0 or 2 (A/B K-row split per lane half)
  const int Mlane = lane & 15;
  const int kern  = (Mlane < td.nch) ? perm[td.c0_rank + Mlane] : -1;

  // A matrix (16ch x 4taps): VGPR0 = K=0/K=2, VGPR1 = K=1/K=3 per lane half
  v2f A[NBLK];
#pragma unroll
  for (int blk = 0; blk < NBLK; blk++) {
    int j0 = blk * 4 + kband;
    float a0 = 0.f, a1 = 0.f;
    if (kern >= 0) {
      if (j0     < td.k) a0 = Wt[kern * KMAX + j0];
      if (j0 + 1 < td.k) a1 = Wt[kern * KMAX + j0 + 1];
    }
    A[blk].x = a0; A[blk].y = a1;
  }

  // bias-seeded accumulator + per-row output pointers (C/D: VGPR j -> M=j / j+8)
  v8f cinit;
  float* rowptr[8];
#pragma unroll
  for (int j = 0; j < 8; j++) {
    int M = j + ((lane >> 4) << 3);
    cinit[j] = (M < td.nch) ? Bv[perm[td.c0_rank + M]] : 0.f;
    rowptr[j] = out + (outbase + (long long)(td.c_local0 + M) * td.Lg + tcol);
  }

  // per-lane signal offsets for b.x per block (b.y = +d)
  int offx[NBLK];
#pragma unroll
  for (int blk = 0; blk < NBLK; blk++)
    offx[blk] = tcol - td.p + (blk * 4 + kband) * td.d;

  const int span = (td.k - 1) * td.d;   // last tap offset
  const bool fullch = (td.nch == 16);

  for (int tt = t_begin; tt < t_end; tt += 16) {
    // wave-uniform interior test -> scalar branch, no EXEC churn
    const bool interior = (tt - td.p >= 0) && (tt + 15 - td.p + span < SL) &&
                          (tt + 16 <= td.Lg) && fullch;
    v8f acc = cinit;
    if (interior) {
      v2f b[NBLK];
#pragma unroll
      for (int blk = 0; blk < NBLK; blk++) {        // batched, unconditional loads
        b[blk].x = xs[tt + offx[blk]];
        b[blk].y = xs[tt + offx[blk] + td.d];
      }
#pragma unroll
      for (int blk = 0; blk < NBLK; blk++)
        acc = __builtin_amdgcn_wmma_f32_16x16x4_f32(false, A[blk], false, b[blk],
                                                    (short)0, acc, false, false);
#pragma unroll
      for (int j = 0; j < 8; j++)                   // unconditional NT stores
        __builtin_nontemporal_store(acc[j], rowptr[j] + tt);
    } else {
      v2f b[NBLK];
#pragma unroll
      for (int blk = 0; blk < NBLK; blk++) {        // branchless clamp + select
        int ix = tt + offx[blk];
        int iy = ix + td.d;
        int cx = min(max(ix, 0), SL - 1);
        int cy = min(max(iy, 0), SL - 1);
        float vx = xs[cx];
        float vy = xs[cy];
        b[blk].x = (ix == cx) ? vx : 0.f;
        b[blk].y = (iy == cy) ? vy : 0.f;
      }
#pragma unroll
      for (int blk = 0; blk < NBLK; blk++)
        acc = __builtin_amdgcn_wmma_f32_16x16x4_f32(false, A[blk], false, b[blk],
                                                    (short)0, acc, false, false);
      const bool tvalid = (tt + tcol < td.Lg);
#pragma unroll
      for (int j = 0; j < 8; j++) {
        int M = j + ((lane >> 4) << 3);
        if (tvalid && M < td.nch)
          __builtin_nontemporal_store(acc[j], rowptr[j] + tt);
      }
    }
  }
}

__global__ __launch_bounds__(256) void rocket_conv(const float* __restrict__ x,
                                                   const float* __restrict__ Wt,
                                                   const float* __restrict__ Bv,
                                                   float* __restrict__ out,
                                                   const int* __restrict__ ws) {
  const int lane = threadIdx.x & 31;
  const int wid  = blockIdx.x * (blockDim.x >> 5) + (threadIdx.x >> 5);
  const int total_work = ws[1];
  if (wid >= total_work) return;   // wave-uniform exit: EXEC all-1s for WMMA

  const int* perm = ws + 16;
  const TileDesc* tiles = (const TileDesc*)(ws + 16 + NK);

  const int tidx  = wid / (NS * TCHUNKS);
  const int rem   = wid % (NS * TCHUNKS);
  const int n     = rem / TCHUNKS;
  const int chunk = rem % TCHUNKS;
  const TileDesc td = tiles[tidx];

  const float* xs = x + n * SL;
  const long long outbase = td.base + (long long)n * td.Cg * td.Lg;

  const int ntiles_t = (td.Lg + 15) >> 4;
  const int tpc = (ntiles_t + TCHUNKS - 1) / TCHUNKS;
  int t_begin = chunk * tpc * 16;
  int t_end   = (chunk + 1) * tpc * 16;
  if (t_end > td.Lg) t_end = td.Lg;

  if (td.k == 7)
    conv_run<2>(td, perm, xs, Wt, Bv, out, outbase, t_begin, t_end, lane);
  else
    conv_run<3>(td, perm, xs, Wt, Bv, out, outbase, t_begin, t_end, lane);
}

// ---------------- host launcher ----------------
extern "C" void kernel_launch(void* const* d_in, const int* in_sizes, int n_in,
                              void* d_out, int out_size, void* d_ws, size_t ws_size,
                              hipStream_t stream) {
  const float* x  = (const float*)d_in[0];  // signal  [2, 32768]
  const float* Wt = (const float*)d_in[1];  // weights [2000, 11]
  const float* Bv = (const float*)d_in[2];  // biases  [2000]
  float* out = (float*)d_out;
  int* ws = (int*)d_ws;

  hipLaunchKernelGGL(rocket_setup, dim3(1), dim3(256), 0, stream, ws);

  const int max_waves = MAX_TILES * NS * TCHUNKS;   // 16384
  const int blocks = max_waves / 8;                 // 8 waves (256 thr) per block
  hipLaunchKernelGGL(rocket_conv, dim3(blocks), dim3(256), 0, stream,
                     x, Wt, Bv, out, ws);
}